// FlexibleGatedDynamicDecoupler_71571335020885
// MI455X (gfx1250) — compile-verified
//
#include <hip/hip_runtime.h>
#include <hip/hip_bf16.h>

typedef __attribute__((ext_vector_type(2))) float v2f;
typedef __attribute__((ext_vector_type(8))) float v8f;

#define BATCH 16
#define CHAN  67
#define HH    128
#define WW    128
#define PM    72   // main filter params (G*K2)
#define PR    9    // remainder filter params
#define EPS   1e-5f

__device__ __forceinline__ float sigmoidf_(float v) { return 1.0f / (1.0f + __expf(-v)); }

// ---------------------------------------------------------------------------
// Kernel 1: global average pool  x(B,C,H,W) -> pooled(B*C)
// ---------------------------------------------------------------------------
__global__ __launch_bounds__(256)
void pool_kernel(const float* __restrict__ x, float* __restrict__ pooled) {
    const int bc = blockIdx.x;                       // 0 .. B*C-1
    const float4* p = (const float4*)(x + (size_t)bc * (HH * WW));
    float s = 0.0f;
    for (int i = threadIdx.x; i < (HH * WW) / 4; i += 256) {
        float4 v = p[i];
        s += v.x + v.y + v.z + v.w;
    }
    // wave32 reduce
    for (int off = 16; off > 0; off >>= 1) s += __shfl_down(s, off, 32);
    __shared__ float ls[8];
    if ((threadIdx.x & 31) == 0) ls[threadIdx.x >> 5] = s;
    __syncthreads();
    if (threadIdx.x == 0) {
        float t = 0.0f;
        #pragma unroll
        for (int i = 0; i < 8; ++i) t += ls[i];
        pooled[bc] = t * (1.0f / (HH * WW));
    }
}

// ---------------------------------------------------------------------------
// Kernel 2: dynamic kernel generation (one wave32, WMMA f32 16x16x4)
//   raw = pooled @ W^T ; gate = raw @ Wg^T ; v = raw*sigmoid(gate) ; BN ; softmax
//
// All operands staged in zero-padded LDS so the WMMA inner loops are
// branchless ds_load_b64 pairs + v_wmma (no exec-mask dances around vmem).
// A (16x4 f32): lanes 0-15 M=lane {K=k0,k0+1}; lanes 16-31 M=lane-16 {K=k0+2,k0+3}
// B (4x16 f32): transpose-symmetric: lane half selects K pair, lid = N
// C/D (16x16 f32): VGPR r, lane<16 -> (M=r, N=lane); lane>=16 -> (M=r+8)
// N-overhang (72->80) handled by clamping the B row index: clamped lanes only
// produce D columns >= 72, which are never read downstream.
// ---------------------------------------------------------------------------
__global__ __launch_bounds__(32)
void gen_kernels(const float* __restrict__ pooled,       // 16*67
                 const float* __restrict__ w_main,       // 72*67
                 const float* __restrict__ w_gate_main,  // 72*72
                 const float* __restrict__ w_rem,        // 9*67
                 const float* __restrict__ w_gate_rem,   // 9*9
                 const float* __restrict__ g_m, const float* __restrict__ be_m,
                 const float* __restrict__ mu_m, const float* __restrict__ va_m,
                 const float* __restrict__ g_r, const float* __restrict__ be_r,
                 const float* __restrict__ mu_r, const float* __restrict__ va_r,
                 float* __restrict__ km_out,             // 16*72 (softmaxed)
                 float* __restrict__ kr_out)             // 16*9  (softmaxed)
{
    __shared__ float sp[16][68];    // pooled, col 67 zero
    __shared__ float rm[16][80];    // main raw -> bn'd values
    __shared__ float gm[16][80];    // main gate logits
    __shared__ float rr[16][16];    // rem raw -> bn'd values (cols 9..15 zero)
    __shared__ float gr[16][16];    // rem gate logits
    __shared__ float wmL[72][68];   // w_main, col 67 zero
    __shared__ float wgL[72][72];   // w_gate_main (K=72 exact, no pad)
    __shared__ float wrL[16][68];   // w_rem, rows 9..15 & col 67 zero
    __shared__ float wgrL[16][12];  // w_gate_rem, rows 9..15 & cols 9..11 zero

    const int lane = threadIdx.x;
    const int half = lane >> 4;     // 0: K pair {k,k+1}, 1: K pair {k+2,k+3}
    const int lid  = lane & 15;
    const int koff = 2 * half;

    // ---- zero-fill padded buffers ----
    for (int i = lane; i < 16 * 68; i += 32) ((float*)sp)[i]   = 0.0f;
    for (int i = lane; i < 72 * 68; i += 32) ((float*)wmL)[i]  = 0.0f;
    for (int i = lane; i < 16 * 68; i += 32) ((float*)wrL)[i]  = 0.0f;
    for (int i = lane; i < 16 * 12; i += 32) ((float*)wgrL)[i] = 0.0f;
    __syncthreads();

    // ---- stage operands into LDS (rectangular copies, no conditionals) ----
    for (int i = lane; i < 16 * 67; i += 32) sp [i / 67][i % 67] = pooled[i];
    for (int i = lane; i < 72 * 67; i += 32) wmL[i / 67][i % 67] = w_main[i];
    for (int i = lane; i < 72 * 72; i += 32) wgL[i / 72][i % 72] = w_gate_main[i];
    for (int i = lane; i < PR * 67; i += 32) wrL[i / 67][i % 67] = w_rem[i];
    for (int i = lane; i < PR * PR; i += 32) wgrL[i / PR][i % PR] = w_gate_rem[i];
    __syncthreads();

    // ---- main GEMM1: rm = pooled @ w_main^T  (16 x 72, padded to 80) ----
    for (int nt = 0; nt < 5; ++nt) {
        const int n = min(nt * 16 + lid, 71);   // clamp: cols >=72 of D unused
        v8f acc = {};
        for (int kc = 0; kc < 17; ++kc) {
            const int k = kc * 4 + koff;
            v2f a = *(const v2f*)&sp[lid][k];
            v2f b = *(const v2f*)&wmL[n][k];
            acc = __builtin_amdgcn_wmma_f32_16x16x4_f32(false, a, false, b,
                                                        (short)0, acc, false, false);
        }
        #pragma unroll
        for (int r = 0; r < 8; ++r) rm[r + 8 * half][nt * 16 + lid] = acc[r];
    }
    __syncthreads();

    // ---- main GEMM2: gm = rm @ w_gate_main^T  (K = 72 exact) ----
    for (int nt = 0; nt < 5; ++nt) {
        const int n = min(nt * 16 + lid, 71);
        v8f acc = {};
        for (int kc = 0; kc < 18; ++kc) {
            const int k = kc * 4 + koff;
            v2f a = *(const v2f*)&rm[lid][k];
            v2f b = *(const v2f*)&wgL[n][k];
            acc = __builtin_amdgcn_wmma_f32_16x16x4_f32(false, a, false, b,
                                                        (short)0, acc, false, false);
        }
        #pragma unroll
        for (int r = 0; r < 8; ++r) gm[r + 8 * half][nt * 16 + lid] = acc[r];
    }
    __syncthreads();

    // ---- rem GEMM1: rr = pooled @ w_rem^T  (16 x 9, padded to 16) ----
    {
        v8f acc = {};
        for (int kc = 0; kc < 17; ++kc) {
            const int k = kc * 4 + koff;
            v2f a = *(const v2f*)&sp[lid][k];
            v2f b = *(const v2f*)&wrL[lid][k];   // rows 9..15 zero -> rr cols 9..15 == 0
            acc = __builtin_amdgcn_wmma_f32_16x16x4_f32(false, a, false, b,
                                                        (short)0, acc, false, false);
        }
        #pragma unroll
        for (int r = 0; r < 8; ++r) rr[r + 8 * half][lid] = acc[r];
    }
    __syncthreads();

    // ---- rem GEMM2: gr = rr @ w_gate_rem^T  (K = 9 -> 3 chunks, zero-padded) ----
    {
        v8f acc = {};
        for (int kc = 0; kc < 3; ++kc) {
            const int k = kc * 4 + koff;
            v2f a = *(const v2f*)&rr[lid][k];
            v2f b = *(const v2f*)&wgrL[lid][k];  // rows 9..15 & cols 9..11 zero
            acc = __builtin_amdgcn_wmma_f32_16x16x4_f32(false, a, false, b,
                                                        (short)0, acc, false, false);
        }
        #pragma unroll
        for (int r = 0; r < 8; ++r) gr[r + 8 * half][lid] = acc[r];
    }
    __syncthreads();

    // ---- elementwise: gate (sigmoid) + BatchNorm ----
    for (int i = lane; i < 16 * PM; i += 32) {
        int b = i / PM, p = i % PM;
        float v = rm[b][p] * sigmoidf_(gm[b][p]);
        rm[b][p] = g_m[p] * (v - mu_m[p]) * rsqrtf(va_m[p] + EPS) + be_m[p];
    }
    for (int i = lane; i < 16 * PR; i += 32) {
        int b = i / PR, p = i % PR;
        float v = rr[b][p] * sigmoidf_(gr[b][p]);
        rr[b][p] = g_r[p] * (v - mu_r[p]) * rsqrtf(va_r[p] + EPS) + be_r[p];
    }
    __syncthreads();

    // ---- softmax over K2=9: main (128 rows of 9) ----
    for (int row = lane; row < 16 * 8; row += 32) {
        int b = row >> 3, g = row & 7;
        const float* rp = &rm[b][g * 9];
        float mx = rp[0];
        #pragma unroll
        for (int k = 1; k < 9; ++k) mx = fmaxf(mx, rp[k]);
        float e[9], s = 0.0f;
        #pragma unroll
        for (int k = 0; k < 9; ++k) { e[k] = __expf(rp[k] - mx); s += e[k]; }
        float inv = 1.0f / s;
        #pragma unroll
        for (int k = 0; k < 9; ++k) km_out[b * PM + g * 9 + k] = e[k] * inv;
    }
    // ---- softmax rem (16 rows of 9) ----
    if (lane < 16) {
        const float* rp = &rr[lane][0];
        float mx = rp[0];
        #pragma unroll
        for (int k = 1; k < 9; ++k) mx = fmaxf(mx, rp[k]);
        float e[9], s = 0.0f;
        #pragma unroll
        for (int k = 0; k < 9; ++k) { e[k] = __expf(rp[k] - mx); s += e[k]; }
        float inv = 1.0f / s;
        #pragma unroll
        for (int k = 0; k < 9; ++k) kr_out[lane * PR + k] = e[k] * inv;
    }
}

// ---------------------------------------------------------------------------
// Kernel 3: dynamic 3x3 conv (reflect pad) + residual. Bandwidth-bound.
// grid = (8 row-strips, 67 channels, 16 batch), 256 threads, LDS tile 18x130.
// ---------------------------------------------------------------------------
#define TILE_H 16
__global__ __launch_bounds__(256)
void conv_kernel(const float* __restrict__ x,
                 const float* __restrict__ km,   // 16*72
                 const float* __restrict__ kr,   // 16*9
                 float* __restrict__ out,
                 float* __restrict__ out_high)
{
    const int strip = blockIdx.x;   // 0..7
    const int c     = blockIdx.y;   // 0..66
    const int b     = blockIdx.z;   // 0..15

    __shared__ float tile[TILE_H + 2][WW + 2];

    const size_t plane = ((size_t)b * CHAN + c) * (HH * WW);
    const float* xs = x + plane;

    // per-(b,c) dynamic 3x3 weights (blockIdx-uniform -> scalar loads)
    float wk[9];
    if (c < 64) {
        const int g = c >> 3;
        #pragma unroll
        for (int k = 0; k < 9; ++k) wk[k] = km[b * PM + g * 9 + k];
    } else {
        #pragma unroll
        for (int k = 0; k < 9; ++k) wk[k] = kr[b * PR + k];
    }

    const int h0 = strip * TILE_H;
    for (int i = threadIdx.x; i < (TILE_H + 2) * (WW + 2); i += 256) {
        int r   = i / (WW + 2);
        int col = i % (WW + 2);
        int gh  = h0 + r - 1;
        int gw  = col - 1;
        gh = (gh < 0) ? -gh : ((gh >= HH) ? (2 * HH - 2 - gh) : gh);   // reflect (no edge repeat)
        gw = (gw < 0) ? -gw : ((gw >= WW) ? (2 * WW - 2 - gw) : gw);
        tile[r][col] = xs[gh * WW + gw];
    }
    __syncthreads();

    const int tcol = (threadIdx.x & 31) * 4;   // 0..124, 4 adjacent columns
    const int trow = threadIdx.x >> 5;         // 0..7
    float* op = out      + plane;
    float* hp = out_high + plane;

    #pragma unroll
    for (int rs = 0; rs < 2; ++rs) {
        const int r = trow + rs * 8;           // 0..15 within strip
        float acc0 = 0.f, acc1 = 0.f, acc2 = 0.f, acc3 = 0.f;
        #pragma unroll
        for (int di = 0; di < 3; ++di) {
            #pragma unroll
            for (int dj = 0; dj < 3; ++dj) {
                const float wv = wk[di * 3 + dj];
                acc0 += wv * tile[r + di][tcol + dj + 0];
                acc1 += wv * tile[r + di][tcol + dj + 1];
                acc2 += wv * tile[r + di][tcol + dj + 2];
                acc3 += wv * tile[r + di][tcol + dj + 3];
            }
        }
        const int h = h0 + r;
        float4 o = { acc0, acc1, acc2, acc3 };
        float4 hq = { tile[r + 1][tcol + 1] - acc0,
                      tile[r + 1][tcol + 2] - acc1,
                      tile[r + 1][tcol + 3] - acc2,
                      tile[r + 1][tcol + 4] - acc3 };
        *(float4*)&op[h * WW + tcol] = o;
        *(float4*)&hp[h * WW + tcol] = hq;
    }
}

// ---------------------------------------------------------------------------
extern "C" void kernel_launch(void* const* d_in, const int* in_sizes, int n_in,
                              void* d_out, int out_size, void* d_ws, size_t ws_size,
                              hipStream_t stream) {
    const float* x           = (const float*)d_in[0];
    const float* w_main      = (const float*)d_in[1];
    const float* w_gate_main = (const float*)d_in[2];
    const float* w_rem       = (const float*)d_in[3];
    const float* w_gate_rem  = (const float*)d_in[4];
    const float* bn_m_gamma  = (const float*)d_in[5];
    const float* bn_m_beta   = (const float*)d_in[6];
    const float* bn_m_mean   = (const float*)d_in[7];
    const float* bn_m_var    = (const float*)d_in[8];
    const float* bn_r_gamma  = (const float*)d_in[9];
    const float* bn_r_beta   = (const float*)d_in[10];
    const float* bn_r_mean   = (const float*)d_in[11];
    const float* bn_r_var    = (const float*)d_in[12];

    float* ws     = (float*)d_ws;
    float* pooled = ws;                       // 16*67  = 1072 floats
    float* km     = ws + 1072;                // 16*72  = 1152 floats
    float* kr     = ws + 1072 + 1152;         // 16*9   = 144 floats

    float* out      = (float*)d_out;
    float* out_high = out + (size_t)BATCH * CHAN * HH * WW;

    pool_kernel<<<dim3(BATCH * CHAN), dim3(256), 0, stream>>>(x, pooled);

    gen_kernels<<<dim3(1), dim3(32), 0, stream>>>(
        pooled, w_main, w_gate_main, w_rem, w_gate_rem,
        bn_m_gamma, bn_m_beta, bn_m_mean, bn_m_var,
        bn_r_gamma, bn_r_beta, bn_r_mean, bn_r_var,
        km, kr);

    conv_kernel<<<dim3(8, CHAN, BATCH), dim3(256), 0, stream>>>(x, km, kr, out, out_high);
}